// Stress2Fringe_85787676770581
// MI455X (gfx1250) — compile-verified
//
#include <hip/hip_runtime.h>

#define HW 50176          // 224*224
#define NB 371            // spectral bands
#define NBP 372           // padded to multiple of 4 (pad weight = 0)
#define REV_CONST 3.5e-5f // 2*pi*H*C/1e-9 / (2*pi)  -> v_cos takes revolutions
#define STRESS_MAG 72000000.0f

typedef float v2f __attribute__((ext_vector_type(2)));
typedef float v8f __attribute__((ext_vector_type(8)));

// ws[0] = stress min bits, ws[1] = stress max bits, ws[2] = iso max bits
__global__ void init_ws_kernel(unsigned int* ws) {
    ws[0] = 0x7F800000u; // +inf
    ws[1] = 0u;          // 0.0f (input is uniform[0,1], non-negative)
    ws[2] = 0u;
}

__global__ void minmax_kernel(const float* __restrict__ in, unsigned int* ws) {
    int tid = blockIdx.x * blockDim.x + threadIdx.x;
    float v = in[tid];            // grid sized to cover HW exactly
    float vmin = v, vmax = v;
    #pragma unroll
    for (int off = 16; off >= 1; off >>= 1) {
        vmin = fminf(vmin, __shfl_xor(vmin, off, 32));
        vmax = fmaxf(vmax, __shfl_xor(vmax, off, 32));
    }
    if ((threadIdx.x & 31) == 0) {
        atomicMin(&ws[0], __float_as_uint(vmin));
        atomicMax(&ws[1], __float_as_uint(vmax));
    }
}

// Each wave computes a 16-pixel tile: D(16x16) += A(16x4) * B(4x16), K looped 0..371.
// A[m,k] = 1 - cos(2*pi * coef_m / lambda_k); B[k,n] = 0.5*interaction[k,n] (n<3), else 0.
__global__ void __launch_bounds__(128)
fringe_wmma_kernel(const float* __restrict__ stress,
                   const float* __restrict__ inter,  // (371,3)
                   float* __restrict__ out,          // (3,224,224) unnormalized
                   unsigned int* ws) {
    __shared__ float ldsW[NBP * 3];
    for (int i = threadIdx.x; i < NBP * 3; i += blockDim.x) {
        int k = i / 3;
        ldsW[i] = (k < NB) ? 0.5f * inter[i] : 0.0f;
    }
    __syncthreads();

    const float smin  = __uint_as_float(ws[0]);
    const float smax  = __uint_as_float(ws[1]);
    const float scale = STRESS_MAG / (smax - smin);

    const int lane = threadIdx.x & 31;
    const int half = lane >> 4;     // 0: lanes 0-15, 1: lanes 16-31
    const int mloc = lane & 15;     // A-matrix row owned by this lane
    const int n    = lane & 15;     // B/C/D column owned by this lane
    const int tile = blockIdx.x * (blockDim.x >> 5) + (threadIdx.x >> 5);
    const int prow = tile * 16 + mloc;

    // revolutions per (1/lambda_nm)
    const float coef = (stress[prow] - smin) * scale * REV_CONST;

    const bool  isw   = (n < 3);
    const float wmask = isw ? 1.0f : 0.0f;

    v8f c = {};
    for (int k0 = 0; k0 < NBP; k0 += 4) {
        const int ka = k0 + 2 * half;            // K index for element v=0 (v=1 is ka+1)
        const float lam0 = (float)(390 + ka);
        const float lam1 = lam0 + 1.0f;
        v2f a, b;
        a.x = 1.0f - __builtin_amdgcn_cosf(coef * __builtin_amdgcn_rcpf(lam0));
        a.y = 1.0f - __builtin_amdgcn_cosf(coef * __builtin_amdgcn_rcpf(lam1));
        const int bi = isw ? (ka * 3 + n) : 0;   // clamped for n>=3 lanes
        b.x = ldsW[bi]     * wmask;
        b.y = ldsW[bi + 3] * wmask;
        // v_wmma_f32_16x16x4_f32: (neg_a, A, neg_b, B, c_mod, C, reuse_a, reuse_b)
        c = __builtin_amdgcn_wmma_f32_16x16x4_f32(false, a, false, b, (short)0, c,
                                                  false, false);
    }

    if (isw) {
        float lmax = 0.0f;
        #pragma unroll
        for (int v = 0; v < 8; ++v) {
            const int m = v + 8 * half;          // D row: VGPR v, lanes 16-31 hold m+8
            const int p = tile * 16 + m;
            const float val = c[v];
            out[n * HW + p] = val;
            lmax = fmaxf(lmax, val);
        }
        atomicMax(&ws[2], __float_as_uint(lmax)); // iso >= 0 -> uint order == float order
    }
}

__global__ void normalize_kernel(float* __restrict__ out, const unsigned int* ws) {
    int i = blockIdx.x * blockDim.x + threadIdx.x;
    if (i < 3 * HW) {
        out[i] = out[i] / __uint_as_float(ws[2]);
    }
}

extern "C" void kernel_launch(void* const* d_in, const int* in_sizes, int n_in,
                              void* d_out, int out_size, void* d_ws, size_t ws_size,
                              hipStream_t stream) {
    const float* stress = (const float*)d_in[0];   // (1,1,224,224) f32
    const float* inter  = (const float*)d_in[1];   // (371,3) f32
    float* out = (float*)d_out;                    // (1,3,224,224) f32
    unsigned int* ws = (unsigned int*)d_ws;

    init_ws_kernel<<<1, 1, 0, stream>>>(ws);
    minmax_kernel<<<HW / 256, 256, 0, stream>>>(stress, ws);   // 196 blocks, 1 elem/thread
    // 3136 tiles of 16 pixels, 4 waves (tiles) per 128-thread block
    fringe_wmma_kernel<<<784, 128, 0, stream>>>(stress, inter, out, ws);
    normalize_kernel<<<(3 * HW + 255) / 256, 256, 0, stream>>>(out, ws);
}